// GatedBlock_87643102642324
// MI455X (gfx1250) — compile-verified
//
#include <hip/hip_runtime.h>
#include <math.h>

// Gated block (e3nn-style): RS = [(256,0),(128,1),(64,2),(32,3)]
// in : (B, 1408) f32   [1184 features | 224 gates]
// out: (B, 1184) f32
//   p in [0,256)      : relu
//   p in [256,640)    : * sigmoid(g[(p-256)/3])
//   p in [640,960)    : * sigmoid(g[128+(p-640)/5])
//   p in [960,1184)   : * sigmoid(g[192+(p-960)/7])
// Pure streaming op: ~679 MB traffic -> HBM-bound (~29us @ 23.3 TB/s).

typedef __attribute__((ext_vector_type(4))) float v4f;
typedef __attribute__((address_space(1))) int  gint_t;   // global
typedef __attribute__((address_space(3))) int  lint_t;   // LDS

constexpr int SIZE_OUT  = 1184;
constexpr int N_GATES   = 224;
constexpr int IN_STRIDE = SIZE_OUT + N_GATES;   // 1408
constexpr int VECS      = SIZE_OUT / 4;         // 296 float4 per row
constexpr int ROWS      = 4;                    // rows per block
constexpr int BLOCK     = 256;                  // 8 wave32 waves

__device__ __forceinline__ float gate_apply(float x, int p, const float* __restrict__ sg) {
  if (p < 256) return fmaxf(x, 0.0f);
  int gi;
  if (p < 640)      gi = (p - 256) / 3;
  else if (p < 960) gi = 128 + (p - 640) / 5;
  else              gi = 192 + (p - 960) / 7;
  return x * sg[gi];
}

__global__ __launch_bounds__(BLOCK) void gated_block_kernel(
    const float* __restrict__ in, float* __restrict__ out, int batch) {
  __shared__ float raw[ROWS * N_GATES];
  __shared__ float sg [ROWS * N_GATES];

  const int t    = threadIdx.x;
  const int row0 = blockIdx.x * ROWS;

  // --- Stage gate tails of ROWS rows into LDS via CDNA5 async DMA path ---
  // (gfx1250 GLOBAL_LOAD_ASYNC_TO_LDS_B32, tracked with ASYNCcnt)
  for (int g = t; g < ROWS * N_GATES; g += BLOCK) {
    const int rl  = g / N_GATES;
    const int gi  = g - rl * N_GATES;
    const int row = row0 + rl;
    if (row < batch) {
      const float* src = in + (size_t)row * IN_STRIDE + SIZE_OUT + gi;
      __builtin_amdgcn_global_load_async_to_lds_b32(
          (gint_t*)src,
          (lint_t*)&raw[g],
          /*imm offset*/0, /*cpol*/0);
    } else {
      raw[g] = 0.0f;
    }
  }
  asm volatile("s_wait_asynccnt 0" ::: "memory");
  __syncthreads();

  // Sigmoid table in LDS (each gate reused 3..7x by the main loop).
  for (int g = t; g < ROWS * N_GATES; g += BLOCK) {
    const float x = raw[g];
    sg[g] = 1.0f / (1.0f + expf(-x));
  }
  __syncthreads();

  // --- Main streaming pass: b128 NT loads/stores, one touch per byte ---
  for (int v = t; v < ROWS * VECS; v += BLOCK) {
    const int rl  = v / VECS;
    const int vec = v - rl * VECS;
    const int row = row0 + rl;
    if (row >= batch) break;   // rl is nondecreasing in v for fixed t

    const v4f* in4 = (const v4f*)(in + (size_t)row * IN_STRIDE);
    v4f x = __builtin_nontemporal_load(in4 + vec);

    const float* s = &sg[rl * N_GATES];
    const int p = vec * 4;
    v4f r;
    r.x = gate_apply(x.x, p + 0, s);
    r.y = gate_apply(x.y, p + 1, s);
    r.z = gate_apply(x.z, p + 2, s);
    r.w = gate_apply(x.w, p + 3, s);

    v4f* out4 = (v4f*)(out + (size_t)row * SIZE_OUT);
    __builtin_nontemporal_store(r, out4 + vec);
  }
}

extern "C" void kernel_launch(void* const* d_in, const int* in_sizes, int n_in,
                              void* d_out, int out_size, void* d_ws, size_t ws_size,
                              hipStream_t stream) {
  const float* in  = (const float*)d_in[0];
  float*       out = (float*)d_out;
  const int batch  = in_sizes[0] / IN_STRIDE;   // 65536 for the reference shapes
  const int grid   = (batch + ROWS - 1) / ROWS; // 16384 blocks
  gated_block_kernel<<<grid, BLOCK, 0, stream>>>(in, out, batch);
}